// iSogCLR_Loss_6493990551848
// MI455X (gfx1250) — compile-verified
//
#include <hip/hip_runtime.h>

#define BSZ 4096
#define DIM 768
#define ROWS_PB 64      // rows per block (2 row-groups of 32)
#define COLS_PC 128     // column-chunk width (4 col-groups of 32)
#define KC 64           // k-chunk staged in LDS
#define AS_STRIDE 68    // padded strides: mult of 4 floats -> float4 aligned, conflict-free frags
#define BS_STRIDE 68
#define TS 132          // spill-tile stride (128 + 4)

typedef float v2f __attribute__((ext_vector_type(2)));
typedef float v8f __attribute__((ext_vector_type(8)));

__device__ __forceinline__ void merge_stats(float& m, float& e, float& es,
                                            float m2, float e2, float es2) {
    float mn = fmaxf(m, m2);
    float f1 = __expf(m - mn);
    float f2 = __expf(m2 - mn);
    e  = e  * f1 + e2  * f2;
    es = es * f1 + es2 * f2;
    m  = mn;
}

// diag[i] = dot(img[i], txt[i]); one wave (32 lanes) per row
__global__ __launch_bounds__(256) void diag_kernel(const float* __restrict__ img,
                                                   const float* __restrict__ txt,
                                                   float* __restrict__ diag) {
    int wave = (blockIdx.x * blockDim.x + threadIdx.x) >> 5;
    int lane = threadIdx.x & 31;
    if (wave >= BSZ) return;
    const float* a = img + (size_t)wave * DIM;
    const float* b = txt + (size_t)wave * DIM;
    float s = 0.f;
    for (int k = lane; k < DIM; k += 32) s += a[k] * b[k];
    #pragma unroll
    for (int off = 16; off > 0; off >>= 1) s += __shfl_down(s, off, 32);
    if (lane == 0) diag[wave] = s;
}

__global__ void gather_tau_kernel(const int* __restrict__ ii, const int* __restrict__ ti,
                                  const float* __restrict__ tau_I, const float* __restrict__ tau_T,
                                  float* __restrict__ tau_img, float* __restrict__ tau_txt) {
    int i = blockIdx.x * blockDim.x + threadIdx.x;
    if (i < BSZ) {
        tau_img[i] = tau_I[ii[i]];
        tau_txt[i] = tau_T[ti[i]];
    }
}

// Fused fp32-WMMA GEMM + online row statistics (flash-softmax style).
// blockIdx.y == 0: image pass (rows of sim);  == 1: text pass (rows of sim^T).
// For each row i: z_j = (A_i . B_j - diag[i]) / tau_row[i]
//   out_m  = running max of z (diagonal contributes z == 0)
//   out_e  = sum_{j != i} exp(z_j - out_m)
//   out_es = sum_{j != i} exp(z_j - out_m) * (s_ij - diag_i)
__global__ __launch_bounds__(256) void stat_kernel(
    const float* __restrict__ img, const float* __restrict__ txt,
    const float* __restrict__ diag,
    const float* __restrict__ tau_img, const float* __restrict__ tau_txt,
    float* __restrict__ mI, float* __restrict__ eI, float* __restrict__ esI,
    float* __restrict__ mT, float* __restrict__ eT, float* __restrict__ esT)
{
    // A/B staging and the spill tile have barrier-separated lifetimes -> alias them.
    __shared__ float smem[ROWS_PB * AS_STRIDE + COLS_PC * BS_STRIDE]; // 13056 floats
    float* As   = smem;                          // 64 x 68
    float* Bs   = smem + ROWS_PB * AS_STRIDE;    // 128 x 68
    float* Tile = smem;                          // 64 x 132 (aliases As/Bs)

    const float* Arow;
    const float* Bcol;
    const float* tau_row;
    float *om, *oe, *oes;
    if (blockIdx.y == 0) { Arow = img; Bcol = txt; tau_row = tau_img; om = mI; oe = eI; oes = esI; }
    else                 { Arow = txt; Bcol = img; tau_row = tau_txt; om = mT; oe = eT; oes = esT; }

    const int t    = threadIdx.x;
    const int lane = t & 31;
    const int w    = t >> 5;      // 8 waves
    const int rg   = w & 1;       // row-group: rows rg*32 .. rg*32+31
    const int cg   = w >> 1;      // col-group: cols cg*32 .. cg*32+31
    const int rowBase = blockIdx.x * ROWS_PB;
    const int fm = lane & 15;     // fragment M/N index
    const int kh = lane >> 4;     // k-half selector for 16x4 / 4x16 fragments

    // reduce-phase role: 4 threads per row, 32 elements each
    const int rrow = t >> 2;              // 0..63
    const int rq   = t & 3;               // column 32-wide slot
    const int gi   = rowBase + rrow;      // global row
    const float dgi     = diag[gi];
    const float inv_tau = 1.0f / tau_row[gi];

    float m_run = 0.0f;   // diagonal term z == 0 participates in the max
    float e_run = 0.0f;
    float es_run = 0.0f;

    const int ar0 = rg * 32 + fm, ar1 = ar0 + 16;
    const int bc0 = cg * 32 + fm, bc1 = bc0 + 16;

    for (int jc = 0; jc < BSZ; jc += COLS_PC) {
        v8f acc00 = {}, acc01 = {}, acc10 = {}, acc11 = {};
        for (int kc = 0; kc < DIM; kc += KC) {
            // stage A rows [rowBase, rowBase+64) k-chunk : 1024 float4, 4 per thread
            #pragma unroll
            for (int idx = t; idx < ROWS_PB * (KC / 4); idx += 256) {
                int r = idx >> 4, q = idx & 15;
                float4 v = *(const float4*)(Arow + (size_t)(rowBase + r) * DIM + kc + q * 4);
                *(float4*)(&As[r * AS_STRIDE + q * 4]) = v;
            }
            // stage B rows (= output columns) [jc, jc+128) k-chunk : 2048 float4, 8/thread
            #pragma unroll
            for (int idx = t; idx < COLS_PC * (KC / 4); idx += 256) {
                int r = idx >> 4, q = idx & 15;
                float4 v = *(const float4*)(Bcol + (size_t)(jc + r) * DIM + kc + q * 4);
                *(float4*)(&Bs[r * BS_STRIDE + q * 4]) = v;
            }
            __syncthreads();
            #pragma unroll
            for (int kb = 0; kb < KC; kb += 4) {
                v2f a0 = *(const v2f*)(&As[ar0 * AS_STRIDE + kb + 2 * kh]);
                v2f a1 = *(const v2f*)(&As[ar1 * AS_STRIDE + kb + 2 * kh]);
                v2f b0 = *(const v2f*)(&Bs[bc0 * BS_STRIDE + kb + 2 * kh]);
                v2f b1 = *(const v2f*)(&Bs[bc1 * BS_STRIDE + kb + 2 * kh]);
                acc00 = __builtin_amdgcn_wmma_f32_16x16x4_f32(false, a0, false, b0, (short)0, acc00, false, false);
                acc01 = __builtin_amdgcn_wmma_f32_16x16x4_f32(false, a0, false, b1, (short)0, acc01, false, false);
                acc10 = __builtin_amdgcn_wmma_f32_16x16x4_f32(false, a1, false, b0, (short)0, acc10, false, false);
                acc11 = __builtin_amdgcn_wmma_f32_16x16x4_f32(false, a1, false, b1, (short)0, acc11, false, false);
            }
            __syncthreads();   // WMMA ds-reads done before Tile overwrites As/Bs
        }
        // C/D layout: lane half selects M base (0/8), VGPR v = M offset, fm = N
        {
            int trow = rg * 32 + kh * 8;
            int tcol = cg * 32 + fm;
            #pragma unroll
            for (int v = 0; v < 8; ++v) {
                Tile[(trow + v)      * TS + tcol]      = acc00[v];
                Tile[(trow + v)      * TS + tcol + 16] = acc01[v];
                Tile[(trow + v + 16) * TS + tcol]      = acc10[v];
                Tile[(trow + v + 16) * TS + tcol + 16] = acc11[v];
            }
        }
        __syncthreads();
        // online row reduction over this 128-column chunk
        {
            float mc = -3.0e38f, ec = 0.f, esc = 0.f;
            float vals[32];
            const float* rp = &Tile[rrow * TS + rq * 32];
            #pragma unroll
            for (int q = 0; q < 8; ++q) {
                float4 v = *(const float4*)(rp + q * 4);
                vals[q * 4 + 0] = v.x; vals[q * 4 + 1] = v.y;
                vals[q * 4 + 2] = v.z; vals[q * 4 + 3] = v.w;
            }
            #pragma unroll
            for (int e2 = 0; e2 < 32; ++e2) {
                int j = jc + rq * 32 + e2;
                if (j == gi) continue;          // mask_neg: skip diagonal
                float z = (vals[e2] - dgi) * inv_tau;
                mc = fmaxf(mc, z);
            }
            #pragma unroll
            for (int e2 = 0; e2 < 32; ++e2) {
                int j = jc + rq * 32 + e2;
                if (j == gi) continue;
                float d = vals[e2] - dgi;
                float ex = __expf(d * inv_tau - mc);
                ec  += ex;
                esc += ex * d;
            }
            // butterfly-merge the 4 threads of this row (same wave)
            #pragma unroll
            for (int off = 1; off < 4; off <<= 1) {
                float m2  = __shfl_xor(mc,  off, 32);
                float e2_ = __shfl_xor(ec,  off, 32);
                float es2 = __shfl_xor(esc, off, 32);
                merge_stats(mc, ec, esc, m2, e2_, es2);
            }
            merge_stats(m_run, e_run, es_run, mc, ec, esc);
        }
        __syncthreads();   // Tile reads done before next chunk's As/Bs staging
    }
    if (rq == 0) {
        om[gi]  = m_run;
        oe[gi]  = e_run;
        oes[gi] = es_run;
    }
}

__global__ __launch_bounds__(256) void finalize_kernel(
    const int* __restrict__ ids,
    const float* __restrict__ s_old, const float* __restrict__ b_old,
    const float* __restrict__ u_old, const float* __restrict__ tau_row,
    const float* __restrict__ m_arr, const float* __restrict__ e_arr,
    const float* __restrict__ es_arr, const int* __restrict__ epoch_p,
    float* __restrict__ s_out, float* __restrict__ b_out,
    float* __restrict__ u_out, float* __restrict__ tau_out,
    float* __restrict__ loss_partial)
{
    __shared__ float red[256];
    int i = blockIdx.x * 256 + threadIdx.x;
    float loss = 0.f;
    if (i < BSZ) {
        int id    = ids[i];
        float ob  = b_old[id];
        float m   = m_arr[i];
        float nb  = fmaxf(m, ob);
        float scl = __expf(m - nb);
        float g   = e_arr[i]  * scl;
        float S1  = es_arr[i] * scl;      // sum exp(idt - nb) * diffs
        float sI  = (epoch_p[0] == 0)
                      ? g
                      : 0.2f * s_old[id] * __expf(ob - nb) + 0.8f * g;
        float sc  = fmaxf(sI, 1e-14f);
        float tau = tau_row[i];
        loss = S1 / sc;                                   // sum w * diffs
        float tw = __logf(sc / (float)(BSZ - 1)) + nb + 8.0f - S1 / (tau * sc);
        tw = fminf(fmaxf(tw, -5.0f), 5.0f);
        float u  = 0.5f * u_old[id] + 0.5f * tw;
        float tb = fminf(fmaxf(tau - 0.03f * u, 0.005f), 0.05f);
        b_out[id]   = nb;
        s_out[id]   = sI;
        u_out[id]   = u;
        tau_out[id] = tb;
    }
    red[threadIdx.x] = loss;
    __syncthreads();
    for (int off = 128; off > 0; off >>= 1) {
        if (threadIdx.x < off) red[threadIdx.x] += red[threadIdx.x + off];
        __syncthreads();
    }
    if (threadIdx.x == 0) loss_partial[blockIdx.x] = red[0];
}

__global__ void loss_kernel(const float* __restrict__ partials, float* __restrict__ out) {
    if (threadIdx.x == 0) {
        float s = 0.f;
        for (int i = 0; i < 32; ++i) s += partials[i];
        out[0] = s / (float)BSZ;   // mean(image_loss) + mean(text_loss)
    }
}

extern "C" void kernel_launch(void* const* d_in, const int* in_sizes, int n_in,
                              void* d_out, int out_size, void* d_ws, size_t ws_size,
                              hipStream_t stream) {
    const float* img   = (const float*)d_in[0];
    const float* txt   = (const float*)d_in[1];
    const int*   iid   = (const int*)d_in[2];
    const int*   tid   = (const int*)d_in[3];
    const int*   epoch = (const int*)d_in[4];
    const float* s_I   = (const float*)d_in[6];
    const float* s_T   = (const float*)d_in[7];
    const float* b_I   = (const float*)d_in[8];
    const float* b_T   = (const float*)d_in[9];
    const float* u_I   = (const float*)d_in[10];
    const float* u_T   = (const float*)d_in[11];
    const float* tau_I = (const float*)d_in[12];
    const float* tau_T = (const float*)d_in[13];
    const size_t N = (size_t)in_sizes[6];

    float* out       = (float*)d_out;
    float* o_loss    = out;
    float* o_tau_img = out + 1;
    float* o_tau_txt = out + 1 + BSZ;
    float* o_sI   = out + 1 + 2 * BSZ;
    float* o_sT   = o_sI   + N;
    float* o_bI   = o_sT   + N;
    float* o_bT   = o_bI   + N;
    float* o_uI   = o_bT   + N;
    float* o_uT   = o_uI   + N;
    float* o_tauI = o_uT   + N;
    float* o_tauT = o_tauI + N;

    float* ws   = (float*)d_ws;
    float* diag = ws;
    float* mI   = ws + BSZ;
    float* eI   = mI + BSZ;
    float* esI  = eI + BSZ;
    float* mT   = esI + BSZ;
    float* eT   = mT + BSZ;
    float* esT  = eT + BSZ;
    float* partials = esT + BSZ;   // 32 floats (img: 0..15, txt: 16..31)

    // seed the big state outputs with the old state (scatter overwrites B entries)
    hipMemcpyAsync(o_sI,   s_I,   N * sizeof(float), hipMemcpyDeviceToDevice, stream);
    hipMemcpyAsync(o_sT,   s_T,   N * sizeof(float), hipMemcpyDeviceToDevice, stream);
    hipMemcpyAsync(o_bI,   b_I,   N * sizeof(float), hipMemcpyDeviceToDevice, stream);
    hipMemcpyAsync(o_bT,   b_T,   N * sizeof(float), hipMemcpyDeviceToDevice, stream);
    hipMemcpyAsync(o_uI,   u_I,   N * sizeof(float), hipMemcpyDeviceToDevice, stream);
    hipMemcpyAsync(o_uT,   u_T,   N * sizeof(float), hipMemcpyDeviceToDevice, stream);
    hipMemcpyAsync(o_tauI, tau_I, N * sizeof(float), hipMemcpyDeviceToDevice, stream);
    hipMemcpyAsync(o_tauT, tau_T, N * sizeof(float), hipMemcpyDeviceToDevice, stream);

    diag_kernel<<<BSZ / 8, 256, 0, stream>>>(img, txt, diag);
    gather_tau_kernel<<<BSZ / 256, 256, 0, stream>>>(iid, tid, tau_I, tau_T,
                                                     o_tau_img, o_tau_txt);

    // fused: y==0 image pass (rows of sim), y==1 text pass (rows of sim^T)
    dim3 sgrid(BSZ / ROWS_PB, 2);
    stat_kernel<<<sgrid, 256, 0, stream>>>(img, txt, diag, o_tau_img, o_tau_txt,
                                           mI, eI, esI, mT, eT, esT);

    finalize_kernel<<<BSZ / 256, 256, 0, stream>>>(iid, s_I, b_I, u_I, o_tau_img,
                                                   mI, eI, esI, epoch,
                                                   o_sI, o_bI, o_uI, o_tauI,
                                                   partials);
    finalize_kernel<<<BSZ / 256, 256, 0, stream>>>(tid, s_T, b_T, u_T, o_tau_txt,
                                                   mT, eT, esT, epoch,
                                                   o_sT, o_bT, o_uT, o_tauT,
                                                   partials + 16);

    loss_kernel<<<1, 32, 0, stream>>>(partials, o_loss);
}